// GravityPooling_60911226192347
// MI455X (gfx1250) — compile-verified
//
#include <hip/hip_runtime.h>
#include <math.h>

// Problem constants (from setup_inputs)
#define BATCH 4
#define NPTS  8192
#define DSIG  64
#define KNN   5
#define NKEEP 2048
#define DELTA 1.0f

// KNN kernel tiling
#define QT 128            // queries per block
#define CT 64             // candidates per LDS d2 tile
#define PITCH 65          // LDS row pitch (65 % 64 == 1 -> conflict-free)

typedef __attribute__((ext_vector_type(2))) float v2f;
typedef __attribute__((ext_vector_type(8))) float v8f;

// ---------------------------------------------------------------------------
// Kernel 1: gravity iterations. One workgroup per batch (covers all 8192 pts),
// so the per-iteration centroid reduction is a plain block reduction.
// ---------------------------------------------------------------------------
__global__ __launch_bounds__(256)
void gp_gravity_kernel(const float* __restrict__ g_coords,
                       const float* __restrict__ t_ptr,
                       const int*   __restrict__ iter_ptr,
                       float* __restrict__ ws_coords) {
    const int b   = blockIdx.x;
    const int tid = threadIdx.x;
    __shared__ float red[256];
    __shared__ float cent[3];

    const float t    = t_ptr[0];
    const float coef = 0.5f * t * t;
    const int iters  = iter_ptr[0];

    const float* src0 = g_coords + (size_t)b * NPTS * 3;
    float*       dst  = ws_coords + (size_t)b * NPTS * 3;

    if (iters == 0) {  // defensive: make ws valid even with zero iterations
        for (int p = tid; p < NPTS; p += 256) {
            dst[p*3+0] = src0[p*3+0];
            dst[p*3+1] = src0[p*3+1];
            dst[p*3+2] = src0[p*3+2];
        }
        return;
    }

    for (int it = 0; it < iters; ++it) {
        const float* src = (it == 0) ? src0 : dst;
        float sx = 0.f, sy = 0.f, sz = 0.f;
        for (int p = tid; p < NPTS; p += 256) {
            sx += src[p*3+0]; sy += src[p*3+1]; sz += src[p*3+2];
        }
        #pragma unroll
        for (int c = 0; c < 3; ++c) {
            float v = (c == 0) ? sx : ((c == 1) ? sy : sz);
            red[tid] = v; __syncthreads();
            for (int s = 128; s > 0; s >>= 1) {
                if (tid < s) red[tid] += red[tid + s];
                __syncthreads();
            }
            if (tid == 0) cent[c] = red[0] * (1.0f / NPTS);
            __syncthreads();
        }
        const float cx = cent[0], cy = cent[1], cz = cent[2];
        for (int p = tid; p < NPTS; p += 256) {
            float x = src[p*3+0], y = src[p*3+1], z = src[p*3+2];
            float dx = cx - x, dy = cy - y, dz = cz - z;
            float dist = sqrtf(dx*dx + dy*dy + dz*dz);
            float s = coef * (dist - DELTA) / dist;
            dst[p*3+0] = x + s * dx;
            dst[p*3+1] = y + s * dy;
            dst[p*3+2] = z + s * dz;
        }
        __syncthreads();
    }
}

// ---------------------------------------------------------------------------
// Kernel 2: fused KNN (top-5 of pairwise d2 via V_WMMA_F32_16X16X4_F32) +
// 5x5 neighborhood density. Block = 128 threads (4 waves), owns 128 queries.
// Candidate tiles of 64 points; 16x16 dot-product tiles computed with WMMA
// (K padded 3->4 with zeros), d2 assembled in registers, staged in LDS,
// scanned row-wise by the owning thread into a private sorted top-5.
// Fragment builds load x/y/z unconditionally then v_cndmask-select per the
// ISA half-wave layout — no exec-mask juggling around the LDS loads.
// ---------------------------------------------------------------------------
__global__ __launch_bounds__(128)
void gp_knn_density_kernel(const float* __restrict__ coords,   // [B][N][3] (ws)
                           int*   __restrict__ nbr,            // [B][N][KNN]
                           float* __restrict__ density) {      // [B][N]
    __shared__ float qx[QT], qy[QT], qz[QT], qsq[QT];
    __shared__ float cxs[CT], cys[CT], czs[CT], csq[CT];
    __shared__ float d2[QT * PITCH];

    const int tid   = threadIdx.x;
    const int lane  = tid & 31;
    const int wave  = tid >> 5;
    const int b     = blockIdx.x >> 6;           // 64 blocks per batch
    const int qbase = (blockIdx.x & 63) * QT;
    const float* cb = coords + (size_t)b * NPTS * 3;

    {   // load this block's 128 query points
        int q = qbase + tid;
        float x = cb[q*3+0], y = cb[q*3+1], z = cb[q*3+2];
        qx[tid] = x; qy[tid] = y; qz[tid] = z; qsq[tid] = x*x + y*y + z*z;
    }

    float td[KNN]; int ti[KNN];
    #pragma unroll
    for (int k = 0; k < KNN; ++k) { td[k] = 3.0e38f; ti[k] = -1; }

    const bool half = (lane >= 16);              // ISA frag half-wave split
    const int  l16  = lane & 15;

    for (int ct = 0; ct < NPTS / CT; ++ct) {
        // Prefetch next candidate tile into cache while this tile is processed
        {
            int nct = (ct + 1 < NPTS / CT) ? (ct + 1) : ct;
            __builtin_prefetch(cb + (size_t)(nct * CT + tid % CT) * 3, 0, 0);
        }
        __syncthreads();
        if (tid < CT) {
            int c = ct * CT + tid;
            float x = cb[c*3+0], y = cb[c*3+1], z = cb[c*3+2];
            cxs[tid] = x; cys[tid] = y; czs[tid] = z; csq[tid] = x*x + y*y + z*z;
        }
        __syncthreads();

        // Each wave: m_tiles {2w, 2w+1} x n_tiles 0..3  => 8 WMMAs / wave / tile
        #pragma unroll
        for (int mi = 0; mi < 2; ++mi) {
            const int m_tile = wave * 2 + mi;
            const int mrow   = m_tile * 16 + l16;
            // A frag (16x4 f32): v0 = K0(lanes0-15)/K2(lanes16-31), v1 = K1/K3(=0)
            const float aqx = qx[mrow], aqy = qy[mrow], aqz = qz[mrow];
            v2f afrag;
            afrag.x = half ? aqz  : aqx;
            afrag.y = half ? 0.0f : aqy;
            const int rbase = m_tile * 16 + (half ? 8 : 0);  // C/D row base
            float sq_m[8];
            #pragma unroll
            for (int r = 0; r < 8; ++r) sq_m[r] = qsq[rbase + r];

            #pragma unroll
            for (int n_tile = 0; n_tile < CT / 16; ++n_tile) {
                const int ncol = n_tile * 16 + l16;
                // B frag (4x16 f32), mirror of A layout
                const float bx = cxs[ncol], by = cys[ncol], bz = czs[ncol];
                v2f bfrag;
                bfrag.x = half ? bz   : bx;
                bfrag.y = half ? 0.0f : by;
                v8f acc = {};
                acc = __builtin_amdgcn_wmma_f32_16x16x4_f32(
                        /*neg_a=*/false, afrag, /*neg_b=*/false, bfrag,
                        /*c_mod=*/(short)0, acc, /*reuse_a=*/false, /*reuse_b=*/false);
                const float sq_n = csq[ncol];
                #pragma unroll
                for (int r = 0; r < 8; ++r) {
                    float v = sq_m[r] + sq_n - 2.0f * acc[r];
                    d2[(rbase + r) * PITCH + ncol] = v;
                }
            }
        }
        __syncthreads();

        // Row scan: thread tid owns query row tid. Strict '<' keeps the
        // lowest index on ties (matches lax.top_k), self (d2~0) lands first.
        const int jbase = ct * CT;
        const float* row = &d2[tid * PITCH];
        #pragma unroll 4
        for (int n = 0; n < CT; ++n) {
            float dv = fmaxf(row[n], 0.0f);
            int j = jbase + n;
            if (dv < td[4]) {
                if (dv < td[3]) {
                    td[4] = td[3]; ti[4] = ti[3];
                    if (dv < td[2]) {
                        td[3] = td[2]; ti[3] = ti[2];
                        if (dv < td[1]) {
                            td[2] = td[1]; ti[2] = ti[1];
                            if (dv < td[0]) {
                                td[1] = td[0]; ti[1] = ti[0];
                                td[0] = dv; ti[0] = j;
                            } else { td[1] = dv; ti[1] = j; }
                        } else { td[2] = dv; ti[2] = j; }
                    } else { td[3] = dv; ti[3] = j; }
                } else { td[4] = dv; ti[4] = j; }
            }
        }
    }

    // Density: sum of the full 5x5 neighborhood distance matrix (= 2 * upper tri)
    const int gi = qbase + tid;
    float nx[KNN], ny[KNN], nz[KNN];
    #pragma unroll
    for (int k = 0; k < KNN; ++k) {
        int j = ti[k];
        nx[k] = cb[j*3+0]; ny[k] = cb[j*3+1]; nz[k] = cb[j*3+2];
        nbr[((size_t)b * NPTS + gi) * KNN + k] = j;
    }
    float dens = 0.0f;
    #pragma unroll
    for (int p = 0; p < KNN; ++p)
        #pragma unroll
        for (int q = p + 1; q < KNN; ++q) {
            float dx = nx[p]-nx[q], dy = ny[p]-ny[q], dz = nz[p]-nz[q];
            dens += sqrtf(dx*dx + dy*dy + dz*dz);
        }
    density[(size_t)b * NPTS + gi] = 2.0f * dens;
}

// ---------------------------------------------------------------------------
// Kernel 3: stable rank-by-counting == argsort(densities)[:2048] semantics.
// Densities (32 KB/batch) cached in LDS; 8 blocks of 1024 threads per batch.
// Threads with rank < NKEEP emit the kept coords and the maintain index.
// ---------------------------------------------------------------------------
__global__ __launch_bounds__(1024)
void gp_rank_select_kernel(const float* __restrict__ density,
                           const float* __restrict__ coords,
                           int*   __restrict__ maintain,
                           float* __restrict__ out_coords) {
    __shared__ float dsh[NPTS];                    // 32 KB
    const int b   = blockIdx.x >> 3;               // 8 blocks per batch
    const int tid = threadIdx.x;
    const float* db = density + (size_t)b * NPTS;
    for (int p = tid; p < NPTS; p += 1024) dsh[p] = db[p];
    __syncthreads();

    const int i = (blockIdx.x & 7) * 1024 + tid;
    const float my = dsh[i];
    int rank = 0;
    for (int j = 0; j < NPTS; ++j) {
        float v = dsh[j];
        rank += (v < my) || (v == my && j < i);    // stable argsort rank
    }
    if (rank < NKEEP) {
        maintain[(size_t)b * NKEEP + rank] = i;
        const float* cbi = coords + ((size_t)b * NPTS + i) * 3;
        float* ob = out_coords + ((size_t)b * NKEEP + rank) * 3;
        ob[0] = cbi[0]; ob[1] = cbi[1]; ob[2] = cbi[2];
    }
}

// ---------------------------------------------------------------------------
// Kernel 4: signal pooling. One 64-thread block per kept vertex; thread t
// averages signal channel t over the 5 neighbors.
// ---------------------------------------------------------------------------
__global__ __launch_bounds__(64)
void gp_pool_signal_kernel(const int*   __restrict__ maintain,
                           const int*   __restrict__ nbr,
                           const float* __restrict__ signal,
                           float* __restrict__ out_sig) {
    const int v = blockIdx.x & (NKEEP - 1);
    const int b = blockIdx.x / NKEEP;
    const int t = threadIdx.x;
    const int i = maintain[(size_t)b * NKEEP + v];
    const int* nb = nbr + ((size_t)b * NPTS + i) * KNN;
    float acc = 0.0f;
    #pragma unroll
    for (int k = 0; k < KNN; ++k) {
        int j = nb[k];
        acc += signal[((size_t)b * NPTS + j) * DSIG + t];
    }
    out_sig[((size_t)b * NKEEP + v) * DSIG + t] = acc * (1.0f / KNN);
}

// ---------------------------------------------------------------------------
extern "C" void kernel_launch(void* const* d_in, const int* in_sizes, int n_in,
                              void* d_out, int out_size, void* d_ws, size_t ws_size,
                              hipStream_t stream) {
    const float* coords_in = (const float*)d_in[0];   // [4,8192,3] f32
    const float* signal    = (const float*)d_in[1];   // [4,8192,64] f32
    const float* t_ptr     = (const float*)d_in[2];   // scalar f32
    const int*   iter_ptr  = (const int*)d_in[3];     // scalar int
    (void)in_sizes; (void)n_in; (void)out_size; (void)ws_size;

    // Workspace layout (floats/ints), ~1.2 MB total
    float* ws_coords   = (float*)d_ws;                         // B*N*3
    float* ws_density  = ws_coords + (size_t)BATCH * NPTS * 3; // B*N
    int*   ws_nbr      = (int*)(ws_density + (size_t)BATCH * NPTS); // B*N*KNN
    int*   ws_maintain = ws_nbr + (size_t)BATCH * NPTS * KNN;  // B*NKEEP

    float* out_coords = (float*)d_out;                          // B*NKEEP*3
    float* out_sig    = out_coords + (size_t)BATCH * NKEEP * 3; // B*NKEEP*DSIG

    gp_gravity_kernel<<<BATCH, 256, 0, stream>>>(coords_in, t_ptr, iter_ptr, ws_coords);
    gp_knn_density_kernel<<<BATCH * (NPTS / QT), QT, 0, stream>>>(ws_coords, ws_nbr, ws_density);
    gp_rank_select_kernel<<<BATCH * (NPTS / 1024), 1024, 0, stream>>>(ws_density, ws_coords,
                                                                      ws_maintain, out_coords);
    gp_pool_signal_kernel<<<BATCH * NKEEP, DSIG, 0, stream>>>(ws_maintain, ws_nbr,
                                                              signal, out_sig);
}